// AttentionDynamicModel_87840671137811
// MI455X (gfx1250) — compile-verified
//
#include <hip/hip_runtime.h>
#include <hip/hip_bf16.h>
#include <stdint.h>

// Problem constants (match reference setup_inputs)
#define BB   256
#define NN   1000
#define DD   128
#define HH   8
#define HDIM 16
#define NT   63        // ceil(NN/16) token tiles
#define NPAD 1008      // NT*16
#define PSTR 1036      // LDS row stride (dwords) for attn weights, bank-conflict padded

typedef __attribute__((ext_vector_type(2))) float v2f;
typedef __attribute__((ext_vector_type(8))) float v8f;

static __device__ __forceinline__ float neg_inf() { return -__builtin_inff(); }

// ---------------------------------------------------------------------------
// Kernel A: per-batch query prep.
//   Q = mean_graph_emb @ wq_context + [E[prev], 1-used] @ wq_step_context
//   q~[h,:] = (wk[:, h*16:(h+1)*16] @ Qh[h]) * (1/sqrt(16))   (heads 8..15 = 0)
// Output qws: [B][16][128] f32 (row = padded head).
// ---------------------------------------------------------------------------
__global__ void __launch_bounds__(128) qprep_kernel(
    const float* __restrict__ E, const float* __restrict__ mge,
    const float* __restrict__ used, const int* __restrict__ prev,
    const float* __restrict__ wq_ctx, const float* __restrict__ wq_step,
    const float* __restrict__ wk, float* __restrict__ qws)
{
    const int b = blockIdx.x;
    const int t = threadIdx.x;           // 0..127
    __shared__ float curL[DD];
    __shared__ float mgeL[DD];
    __shared__ float QL[DD];

    const int pv = prev[b];
    curL[t] = E[((size_t)b * NN + (size_t)pv) * DD + t];
    mgeL[t] = mge[(size_t)b * DD + t];
    __syncthreads();

    // Column t of Q
    float q = (1.0f - used[b]) * wq_step[(size_t)DD * DD + t];   // capacity row (row 128)
    for (int k = 0; k < DD; ++k) {
        q += mgeL[k] * wq_ctx[(size_t)k * DD + t];
        q += curL[k] * wq_step[(size_t)k * DD + t];
    }
    QL[t] = q;
    __syncthreads();

    // q~[h][t] = sum_j wk[t, h*16+j] * Q[h*16+j], scaled by 1/sqrt(hd)=0.25
    float* qb = qws + (size_t)b * (16 * DD);
    const float* wkrow = wk + (size_t)t * DD;
    for (int h = 0; h < HH; ++h) {
        float s = 0.f;
        #pragma unroll
        for (int j = 0; j < HDIM; ++j)
            s += wkrow[h * HDIM + j] * QL[h * HDIM + j];
        qb[h * DD + t]        = 0.25f * s;
        qb[(8 + h) * DD + t]  = 0.0f;       // zero-pad heads 8..15 for 16-row WMMA A
    }
}

// ---------------------------------------------------------------------------
// Kernel B: one workgroup (8 wave32) per batch; fully fused decode step.
// ---------------------------------------------------------------------------
__global__ void __launch_bounds__(256) decode_kernel(
    const float* __restrict__ E, const uint8_t* __restrict__ mask,
    const float* __restrict__ qws, const float* __restrict__ wv,
    const float* __restrict__ w_out, const float* __restrict__ wk_tanh,
    float* __restrict__ out)
{
    const int b    = blockIdx.x;
    const int tid  = threadIdx.x;
    const int wave = tid >> 5;
    const int lane = tid & 31;
    const int g    = lane >> 4;      // half-wave group (0/1)
    const int r    = lane & 15;      // lane-in-group

    __shared__ float pL[HH][PSTR];   // compat -> unnormalized attn weights
    __shared__ float ebarL[HH][DD];  // unnormalized attn @ E
    __shared__ float logitsL[NPAD];
    __shared__ float tmpL[DD];
    __shared__ float mhaL[DD];
    __shared__ float q2L[DD];
    __shared__ float rsumL[HH];
    __shared__ float redL[8];
    __shared__ float bcast[2];

    const float*   Eb = E    + (size_t)b * NN * DD;
    const uint8_t* mb = mask + (size_t)b * NN;

    // -------- preload A operand (q~) registers: A[M=r][K=4k+2g+{0,1}] --------
    const float* qb = qws + (size_t)b * 16 * DD;
    v2f aReg[32];
    #pragma unroll
    for (int k = 0; k < 32; ++k) {
        const float* p = qb + r * DD + 4 * k + 2 * g;
        aReg[k].x = p[0];
        aReg[k].y = p[1];
    }

    // -------- phase 1: compat[h][n] = q~[h] . E[n]  (WMMA f32 16x16x4) --------
    for (int t = wave; t < NT; t += 8) {
        const int ntile = t * 16;
        int tok = ntile + r; if (tok > NN - 1) tok = NN - 1;   // clamp tail
        const float* erow = Eb + (size_t)tok * DD;
        v8f c = {};
        #pragma unroll
        for (int k = 0; k < 32; ++k) {
            v2f bv;                              // B[K=4k+2g+{0,1}][N=r] = E[tok, d]
            bv.x = erow[4 * k + 2 * g];
            bv.y = erow[4 * k + 2 * g + 1];
            c = __builtin_amdgcn_wmma_f32_16x16x4_f32(false, aReg[k], false, bv,
                                                      (short)0, c, false, false);
        }
        #pragma unroll
        for (int gi = 0; gi < 8; ++gi) {
            const int M = gi + 8 * g;            // C layout: VGPR gi, halves = M / M+8
            if (M < HH) pL[M][ntile + r] = c[gi];
        }
    }
    __syncthreads();

    // -------- phase 2: masked softmax weights per head (unnormalized) --------
    {
        const int h = wave;                      // one head per wave
        float mx = neg_inf();
        for (int n = lane; n < NPAD; n += 32) {
            float cv  = pL[h][n];
            bool  bad = (n >= NN) ? true : (mb[n] != 0);
            cv = bad ? neg_inf() : cv;
            pL[h][n] = cv;
            mx = fmaxf(mx, cv);
        }
        #pragma unroll
        for (int o = 16; o > 0; o >>= 1) mx = fmaxf(mx, __shfl_xor(mx, o, 32));
        float s = 0.f;
        for (int n = lane; n < NPAD; n += 32) {
            float cv = pL[h][n];
            float e  = (cv == neg_inf()) ? 0.f : expf(cv - mx);
            pL[h][n] = e;                        // unnormalized weight
            s += e;
        }
        #pragma unroll
        for (int o = 16; o > 0; o >>= 1) s += __shfl_xor(s, o, 32);
        if (lane == 0) rsumL[h] = s;
    }
    __syncthreads();

    // -------- phase 3: ebar[h][d] = sum_n p[h][n] * E[n][d]  (WMMA) --------
    {
        const int dbase = wave * 16;             // each wave owns a 16-wide d slice
        v8f c = {};
        for (int k = 0; k < NPAD / 4; ++k) {     // K = tokens, 252 steps of 4
            const int k0 = 4 * k + 2 * g;
            const int t0 = (k0     < NN) ? k0     : NN - 1;
            const int t1 = (k0 + 1 < NN) ? k0 + 1 : NN - 1;
            v2f av, bv;
            av.x = (r < HH) ? pL[r][k0]     : 0.f;   // A[M=head r][K=token]
            av.y = (r < HH) ? pL[r][k0 + 1] : 0.f;   // (p==0 for n>=NN => clamp safe)
            bv.x = Eb[(size_t)t0 * DD + dbase + r];  // B[K=token][N=d]
            bv.y = Eb[(size_t)t1 * DD + dbase + r];
            c = __builtin_amdgcn_wmma_f32_16x16x4_f32(false, av, false, bv,
                                                      (short)0, c, false, false);
        }
        #pragma unroll
        for (int gi = 0; gi < 8; ++gi) {
            const int M = gi + 8 * g;
            if (M < HH) ebarL[M][dbase + r] = c[gi];
        }
    }
    __syncthreads();

    // -------- phase 4: tmp = (ebar/sum)@wv ; mha = tmp@w_out ; q2 = wk_tanh@mha --------
    if (tid < DD) {
        const int h = tid >> 4;
        const float inv = 1.0f / rsumL[h];
        float s = 0.f;
        for (int d = 0; d < DD; ++d) s += ebarL[h][d] * wv[(size_t)d * DD + tid];
        tmpL[tid] = s * inv;
    }
    __syncthreads();
    if (tid < DD) {
        float s = 0.f;
        for (int k = 0; k < DD; ++k) s += tmpL[k] * w_out[(size_t)k * DD + tid];
        mhaL[tid] = s;
    }
    __syncthreads();
    if (tid < DD) {
        float s = 0.f;
        for (int d = 0; d < DD; ++d) s += wk_tanh[(size_t)tid * DD + d] * mhaL[d];
        q2L[tid] = s * 0.08838834764831845f;     // fold 1/sqrt(D)
    }
    __syncthreads();

    // -------- phase 5: logits[n] = 10*tanh(q2 . E[n]), masked --------
    for (int n = wave; n < NN; n += 8) {
        const float* erow = Eb + (size_t)n * DD;
        float s = 0.f;
        #pragma unroll
        for (int j = 0; j < 4; ++j) s += erow[4 * lane + j] * q2L[4 * lane + j];
        #pragma unroll
        for (int o = 16; o > 0; o >>= 1) s += __shfl_xor(s, o, 32);
        if (lane == 0) {
            const float lg = 10.0f * tanhf(s);
            logitsL[n] = mb[n] ? neg_inf() : lg;
        }
    }
    __syncthreads();

    // -------- phase 6: log_softmax over n --------
    float mx = neg_inf();
    for (int n = tid; n < NN; n += 256) mx = fmaxf(mx, logitsL[n]);
    #pragma unroll
    for (int o = 16; o > 0; o >>= 1) mx = fmaxf(mx, __shfl_xor(mx, o, 32));
    if (lane == 0) redL[wave] = mx;
    __syncthreads();
    if (tid == 0) {
        float m = redL[0];
        for (int w = 1; w < 8; ++w) m = fmaxf(m, redL[w]);
        bcast[0] = m;
    }
    __syncthreads();
    mx = bcast[0];

    float s = 0.f;
    for (int n = tid; n < NN; n += 256) {
        const float l = logitsL[n];
        s += (l == neg_inf()) ? 0.f : expf(l - mx);
    }
    #pragma unroll
    for (int o = 16; o > 0; o >>= 1) s += __shfl_xor(s, o, 32);
    if (lane == 0) redL[wave] = s;
    __syncthreads();
    if (tid == 0) {
        float tot = 0.f;
        for (int w = 0; w < 8; ++w) tot += redL[w];
        bcast[1] = logf(tot);
    }
    __syncthreads();
    const float lse = bcast[1];

    for (int n = tid; n < NN; n += 256)
        out[(size_t)b * NN + n] = logitsL[n] - mx - lse;
}

// ---------------------------------------------------------------------------
extern "C" void kernel_launch(void* const* d_in, const int* in_sizes, int n_in,
                              void* d_out, int out_size, void* d_ws, size_t ws_size,
                              hipStream_t stream)
{
    (void)in_sizes; (void)n_in; (void)out_size; (void)ws_size;
    const float*   E      = (const float*)d_in[0];
    const float*   mge    = (const float*)d_in[1];
    const float*   used   = (const float*)d_in[2];
    const int*     prev   = (const int*)d_in[3];
    const uint8_t* mask   = (const uint8_t*)d_in[4];
    const float*   wq_ctx = (const float*)d_in[5];
    const float*   wq_stp = (const float*)d_in[6];
    const float*   wk     = (const float*)d_in[7];
    const float*   wk_t   = (const float*)d_in[8];
    const float*   wv     = (const float*)d_in[9];
    const float*   w_out  = (const float*)d_in[10];
    float*         out    = (float*)d_out;
    float*         qws    = (float*)d_ws;       // B*16*128 f32 = 2 MB scratch

    qprep_kernel<<<BB, 128, 0, stream>>>(E, mge, used, prev, wq_ctx, wq_stp, wk, qws);
    decode_kernel<<<BB, 256, 0, stream>>>(E, mask, qws, wv, w_out, wk_t, out);
}